// LocalWindowAttentionFusion_49228915147528
// MI455X (gfx1250) — compile-verified
//
#include <hip/hip_runtime.h>

typedef __attribute__((ext_vector_type(2))) float v2f;
typedef __attribute__((ext_vector_type(8))) float v8f;

enum { EP_STORE = 0, EP_ACC = 1, EP_BIAS_RELU = 2, EP_BIAS = 3 };

// CDNA5 async copy global->LDS (ASYNCcnt-tracked) if the toolchain declares it.
#if defined(__has_builtin)
#if __has_builtin(__builtin_amdgcn_global_load_async_to_lds_b32) && \
    __has_builtin(__builtin_amdgcn_s_wait_asynccnt)
#define USE_ASYNC_LDS 1
#endif
#endif
#ifndef USE_ASYNC_LDS
#define USE_ASYNC_LDS 0
#endif

typedef __attribute__((address_space(1))) int gint_t;
typedef __attribute__((address_space(3))) int lint_t;

// ---------------------------------------------------------------------------
// fp32 WMMA GEMM:  Dst[pixel][m] = sum_k A[m][k] * B[k][pixel]  (+ epilogue)
// One workgroup owns a 16-pixel strip and computes ALL M for it, so every
// activation element is read from HBM exactly once; the strip is staged in
// LDS (async copy when available) and K is processed in <=256-wide chunks.
//  B_NHWC=0: B[k][n] = Bsrc[batch*K*Npix + k*Npix + n]   (NCHW input tensor)
//  B_NHWC=1: B[k][n] = Bsrc[(batch*Npix + n)*K + k]      (NHWC workspace)
//  A_T=0:    A[m][k] = A[m*K + k]     A_T=1: A[m][k] = A[k*M + m]
// Dst is NHWC pixel-major with per-pixel stride MC.
// ---------------------------------------------------------------------------
template <int B_NHWC, int A_T, int KC, int MC, int MT>
__global__ __launch_bounds__((MC / (16 * MT)) * 32) void wmma_gemm(
    const float* __restrict__ A, const float* __restrict__ Bsrc,
    const float* __restrict__ bias, float* __restrict__ Dst,
    int Npix, int epilogue)
{
  constexpr int WAVES = MC / (16 * MT);
  constexpr int NT    = WAVES * 32;
  constexpr int KCH   = (KC < 256) ? KC : 256;  // K chunk staged per round
  constexpr int KP    = KCH + 1;                // LDS row pad vs 64 banks

  __shared__ float Bs[16 * KP];

  const int tid   = threadIdx.x;
  const int lane  = tid & 31;
  const int wid   = tid >> 5;
  const int batch = blockIdx.z;
  const int n0    = blockIdx.x * 16;

  // ISA 7.12.2 fp32 fragment layout (wave32):
  //  A 16x4: lanes 0-15 -> K={0,1}, lanes 16-31 -> K={2,3}; M = lane&15
  //  B 4x16: same K split per lane half; N = lane&15
  //  C/D 16x16: VGPR r -> M = r + 8*(lane>=16), N = lane&15
  const int mrow  = lane & 15;
  const int ncol  = lane & 15;
  const int khalf = (lane >> 4) << 1;  // 0 | 2
  const int moff  = (lane >> 4) << 3;  // 0 | 8

  const size_t pixIdx = (size_t)batch * Npix + n0 + ncol;

  v8f acc[MT];
#pragma unroll
  for (int t = 0; t < MT; ++t)
#pragma unroll
    for (int r = 0; r < 8; ++r) acc[t][r] = 0.f;

  if (epilogue == EP_ACC) {
#pragma unroll
    for (int t = 0; t < MT; ++t) {
      const size_t ob = pixIdx * MC + (size_t)((wid * MT + t) * 16 + moff);
#pragma unroll
      for (int r = 0; r < 8; ++r) acc[t][r] = Dst[ob + r];
    }
  }

  // Warm the (L2-resident) weight matrix into near caches: gfx1250 prefetch.
  for (size_t off = (size_t)tid * 256; off < (size_t)MC * KC * 4; off += (size_t)NT * 256)
    __builtin_prefetch((const char*)A + off, 0, 3);

  for (int kc0 = 0; kc0 < KC; kc0 += KCH) {
    // ---- stage B strip chunk (16 pixels x KCH) into LDS: Bs[n*KP + k] ----
    for (int idx = tid; idx < 16 * KCH; idx += NT) {
      int n, k;
      size_t goff;
      if (B_NHWC) {
        n = idx / KCH;
        k = idx - n * KCH;
        goff = ((size_t)batch * Npix + n0 + n) * KC + kc0 + k;
      } else {
        k = idx >> 4;
        n = idx & 15;
        goff = (size_t)batch * KC * (size_t)Npix + (size_t)(kc0 + k) * Npix + n0 + n;
      }
      const int lidx = n * KP + k;
#if USE_ASYNC_LDS
      __builtin_amdgcn_global_load_async_to_lds_b32(
          (gint_t*)(Bsrc + goff), (lint_t*)&Bs[lidx], 0, 0);
#else
      Bs[lidx] = Bsrc[goff];
#endif
    }
#if USE_ASYNC_LDS
    __builtin_amdgcn_s_wait_asynccnt(0);
#endif
    __syncthreads();

    // ---- MT wmma tiles per wave over this K chunk ----
    for (int k0 = 0; k0 < KCH; k0 += 4) {
      const int ka = k0 + khalf;
      const int kg = kc0 + ka;  // global k for A
      v2f b;
      b[0] = Bs[ncol * KP + ka];
      b[1] = Bs[ncol * KP + ka + 1];
#pragma unroll
      for (int t = 0; t < MT; ++t) {
        const int m0 = (wid * MT + t) * 16;
        v2f a;
        if (A_T == 0) {
          const float* ap = A + (size_t)(m0 + mrow) * KC + kg;
          a[0] = ap[0];
          a[1] = ap[1];
        } else {
          a[0] = A[(size_t)kg * MC + m0 + mrow];
          a[1] = A[(size_t)(kg + 1) * MC + m0 + mrow];
        }
        acc[t] = __builtin_amdgcn_wmma_f32_16x16x4_f32(
            /*neg_a=*/false, a, /*neg_b=*/false, b,
            /*c_mod=*/(short)0, acc[t], /*reuse_a=*/false, /*reuse_b=*/false);
      }
    }
    __syncthreads();  // protect Bs before next chunk overwrites it
  }

#pragma unroll
  for (int t = 0; t < MT; ++t) {
    const int mb = (wid * MT + t) * 16 + moff;
    const size_t ob = pixIdx * MC + mb;
    if (epilogue == EP_BIAS_RELU) {
#pragma unroll
      for (int r = 0; r < 8; ++r) {
        float v = acc[t][r] + bias[mb + r];
        Dst[ob + r] = v > 0.f ? v : 0.f;
      }
    } else if (epilogue == EP_BIAS) {
#pragma unroll
      for (int r = 0; r < 8; ++r) Dst[ob + r] = acc[t][r] + bias[mb + r];
    } else {  // EP_STORE / EP_ACC (C folded into acc)
#pragma unroll
      for (int r = 0; r < 8; ++r) Dst[ob + r] = acc[t][r];
    }
  }
}

// ---------------------------------------------------------------------------
// 7x7 local-window attention, one wave32 per (pixel, head); lane = channel.
// Edge padding == coordinate clamp. On may alias Qn (own-Q held in register,
// the 8 heads of a pixel touch disjoint channels).
// ---------------------------------------------------------------------------
__device__ __forceinline__ int clampi(int x, int lo, int hi) {
  return x < lo ? lo : (x > hi ? hi : x);
}

__global__ __launch_bounds__(256) void attn_kernel(
    const float* __restrict__ Qn, const float* __restrict__ Kn,
    const float* __restrict__ Vn, float* __restrict__ On,
    int H, int W, int NpixB)
{
  const int lane = threadIdx.x & 31;
  const int wid  = threadIdx.x >> 5;
  const int ph   = blockIdx.x * 8 + wid;  // pixel*8 + head
  const int pixel = ph >> 3;
  const int head  = ph & 7;
  const int b  = pixel / NpixB;
  const int hw = pixel - b * NpixB;
  const int h = hw / W;
  const int w = hw - h * W;
  const int c = head * 32 + lane;

  const float q = Qn[(size_t)pixel * 256 + c];

  float lg[49];
  int idx = 0;
#pragma unroll
  for (int i = 0; i < 7; ++i) {
    const int hh = clampi(h + i - 3, 0, H - 1);
#pragma unroll
    for (int j = 0; j < 7; ++j) {
      const int ww = clampi(w + j - 3, 0, W - 1);
      const size_t np = ((size_t)b * NpixB + (size_t)hh * W + ww) * 256 + c;
      float p = q * Kn[np];
#pragma unroll
      for (int s = 16; s > 0; s >>= 1) p += __shfl_xor(p, s, 32);
      lg[idx++] = p * 0.17677669529663689f;  // 32^-0.5
    }
  }

  float mx = lg[0];
#pragma unroll
  for (int k = 1; k < 49; ++k) mx = fmaxf(mx, lg[k]);
  float se = 0.f;
#pragma unroll
  for (int k = 0; k < 49; ++k) { lg[k] = __expf(lg[k] - mx); se += lg[k]; }
  const float inv = 1.f / se;

  float o = 0.f;
  idx = 0;
#pragma unroll
  for (int i = 0; i < 7; ++i) {
    const int hh = clampi(h + i - 3, 0, H - 1);
#pragma unroll
    for (int j = 0; j < 7; ++j) {
      const int ww = clampi(w + j - 3, 0, W - 1);
      const size_t np = ((size_t)b * NpixB + (size_t)hh * W + ww) * 256 + c;
      o += lg[idx++] * inv * Vn[np];
    }
  }
  On[(size_t)pixel * 256 + c] = o;
}

// ---------------------------------------------------------------------------
// LayerNorm over 256 channels, one wave per pixel (8 channels/lane), in place.
// ---------------------------------------------------------------------------
__global__ __launch_bounds__(256) void ln_kernel(
    float* __restrict__ X, const float* __restrict__ g,
    const float* __restrict__ bta, long npix)
{
  const int lane = threadIdx.x & 31;
  const int wid  = threadIdx.x >> 5;
  const long p = (long)blockIdx.x * 8 + wid;
  if (p >= npix) return;

  float v[8];
  float s = 0.f;
#pragma unroll
  for (int r = 0; r < 8; ++r) {
    v[r] = X[p * 256 + lane * 8 + r];
    s += v[r];
  }
#pragma unroll
  for (int t = 16; t > 0; t >>= 1) s += __shfl_xor(s, t, 32);
  const float mu = s * (1.f / 256.f);

  float sq = 0.f;
#pragma unroll
  for (int r = 0; r < 8; ++r) { const float d = v[r] - mu; sq += d * d; }
#pragma unroll
  for (int t = 16; t > 0; t >>= 1) sq += __shfl_xor(sq, t, 32);
  const float rstd = rsqrtf(sq * (1.f / 256.f) + 1e-5f);

#pragma unroll
  for (int r = 0; r < 8; ++r) {
    const int c = lane * 8 + r;
    X[p * 256 + c] = (v[r] - mu) * rstd * g[c] + bta[c];
  }
}

// ---------------------------------------------------------------------------
// Final LN of (x + y) + fused NHWC -> NCHW transpose into d_out.
// ---------------------------------------------------------------------------
__global__ __launch_bounds__(256) void ln2_out_kernel(
    const float* __restrict__ X, const float* __restrict__ Y,
    const float* __restrict__ g, const float* __restrict__ bta,
    float* __restrict__ Out, int NpixB)
{
  const int lane = threadIdx.x & 31;
  const int wid  = threadIdx.x >> 5;
  const long p = (long)blockIdx.x * 8 + wid;
  const int b  = (int)(p / NpixB);
  const int hw = (int)(p - (long)b * NpixB);

  float v[8];
  float s = 0.f;
#pragma unroll
  for (int r = 0; r < 8; ++r) {
    const size_t o = p * 256 + lane * 8 + r;
    v[r] = X[o] + Y[o];
    s += v[r];
  }
#pragma unroll
  for (int t = 16; t > 0; t >>= 1) s += __shfl_xor(s, t, 32);
  const float mu = s * (1.f / 256.f);

  float sq = 0.f;
#pragma unroll
  for (int r = 0; r < 8; ++r) { const float d = v[r] - mu; sq += d * d; }
#pragma unroll
  for (int t = 16; t > 0; t >>= 1) sq += __shfl_xor(sq, t, 32);
  const float rstd = rsqrtf(sq * (1.f / 256.f) + 1e-5f);

#pragma unroll
  for (int r = 0; r < 8; ++r) {
    const int c = lane * 8 + r;
    Out[((size_t)b * 256 + c) * NpixB + hw] = (v[r] - mu) * rstd * g[c] + bta[c];
  }
}

// ---------------------------------------------------------------------------
extern "C" void kernel_launch(void* const* d_in, const int* in_sizes, int n_in,
                              void* d_out, int out_size, void* d_ws, size_t ws_size,
                              hipStream_t stream) {
  (void)in_sizes; (void)n_in; (void)out_size; (void)ws_size;

  const float* F_lidar = (const float*)d_in[0];   // (2,64,200,200)
  const float* F_cam   = (const float*)d_in[1];   // (2,256,200,200)
  const float* Wq = (const float*)d_in[2];        // (256,64)
  const float* Wk = (const float*)d_in[3];        // (256,256)
  const float* Wv = (const float*)d_in[4];        // (256,256)
  const float* Wo = (const float*)d_in[5];        // (256,256)
  const float* Wr = (const float*)d_in[6];        // (256,64)
  const float* g1 = (const float*)d_in[7];
  const float* b1 = (const float*)d_in[8];
  const float* g2 = (const float*)d_in[9];
  const float* b2 = (const float*)d_in[10];
  const float* W1  = (const float*)d_in[11];      // (256,1024) K-major
  const float* bf1 = (const float*)d_in[12];
  const float* W2  = (const float*)d_in[13];      // (1024,256) K-major
  const float* bf2 = (const float*)d_in[14];
  float* Out = (float*)d_out;

  const int H = 200, W = 200, NpixB = 40000;
  const size_t NP = 2ull * NpixB;  // 80000 pixels total

  // Workspace layout (NHWC pixel-major fp32)
  float* Kn  = (float*)d_ws;           // NP*256
  float* Vn  = Kn + NP * 256;          // NP*256
  float* Qn  = Vn + NP * 256;          // NP*256  (attention writes in place)
  float* Xb  = Qn + NP * 256;          // NP*256
  float* Yb  = Xb + NP * 256;          // NP*256
  float* Hid = Yb + NP * 256;          // NP*1024

  const dim3 grid(NpixB / 16, 1, 2);   // 2500 pixel strips x 2 batches

  // Projections: Q = Wq*F_lidar, K = Wk*F_cam, V = Wv*F_cam
  wmma_gemm<0, 0, 64,  256, 4><<<grid, 128, 0, stream>>>(Wq, F_lidar, nullptr, Qn, NpixB, EP_STORE);
  wmma_gemm<0, 0, 256, 256, 4><<<grid, 128, 0, stream>>>(Wk, F_cam,   nullptr, Kn, NpixB, EP_STORE);
  wmma_gemm<0, 0, 256, 256, 4><<<grid, 128, 0, stream>>>(Wv, F_cam,   nullptr, Vn, NpixB, EP_STORE);

  // Local-window attention (in-place over Qn): 80000 px * 8 heads, 8 waves/block
  attn_kernel<<<(int)NP, 256, 0, stream>>>(Qn, Kn, Vn, Qn, H, W, NpixB);

  // x = Wo*attn_out + Wr*F_lidar, then LayerNorm1
  wmma_gemm<1, 0, 256, 256, 4><<<grid, 128, 0, stream>>>(Wo, Qn,      nullptr, Xb, NpixB, EP_STORE);
  wmma_gemm<0, 0, 64,  256, 4><<<grid, 128, 0, stream>>>(Wr, F_lidar, nullptr, Xb, NpixB, EP_ACC);
  ln_kernel<<<(int)(NP / 8), 256, 0, stream>>>(Xb, g1, b1, (long)NP);

  // FFN: Hid = relu(x@W1 + bf1); y = Hid@W2 + bf2
  wmma_gemm<1, 1, 256,  1024, 8><<<grid, 256, 0, stream>>>(W1, Xb,  bf1, Hid, NpixB, EP_BIAS_RELU);
  wmma_gemm<1, 1, 1024, 256,  4><<<grid, 128, 0, stream>>>(W2, Hid, bf2, Yb,  NpixB, EP_BIAS);

  // LayerNorm2(x + y) + NHWC->NCHW transpose into output
  ln2_out_kernel<<<(int)(NP / 8), 256, 0, stream>>>(Xb, Yb, g2, b2, Out, NpixB);
}